// SpectralConv2d_fast_80178449481969
// MI455X (gfx1250) — compile-verified
//
#include <hip/hip_runtime.h>
#include <hip/hip_bf16.h>

// SpectralConv2d (FNO) for MI455X / gfx1250.
// B=16, Cin=Cout=32, H=W=256, modes 12x12 (two kx corners -> 24 kx rows).
//
// Pipeline (dense partial-DFT GEMMs on v_wmma_f32_16x16x4_f32):
//   K1: x[b,i,256,256] -> x_ft[b,i,24,12] (complex)   [WMMA row-DFT + column reduction]
//   K2: out_ft[b,o,m] = sum_i x_ft[b,i,m] * W[i,o,m]  [complex GEMM via 4 real WMMAs]
//   K3: out_ft -> out[b,o,256,256]                    [T(256x24) @ G(24x256) WMMA]
//
// Memory-bound problem: ~256 MB traffic -> ~11 us at 23.3 TB/s.  Global<->LDS
// movement uses the CDNA5 async engine (ASYNCcnt) double-buffered against the
// WMMA pipeline where available.

typedef __attribute__((ext_vector_type(2))) float v2f;
typedef __attribute__((ext_vector_type(8))) float v8f;

#define WMMA_F32(A, B, C) \
  __builtin_amdgcn_wmma_f32_16x16x4_f32(false, (A), false, (B), (short)0, (C), false, false)

// ---- CDNA5 async global<->LDS path (guarded; sync fallback kept) -----------
#if defined(__has_builtin)
#if __has_builtin(__builtin_amdgcn_global_load_async_to_lds_b128) && \
    __has_builtin(__builtin_amdgcn_global_store_async_from_lds_b128)
#define USE_ASYNC_LDS 1
#endif
#endif
#ifndef USE_ASYNC_LDS
#define USE_ASYNC_LDS 0
#endif

#if USE_ASYNC_LDS
typedef int v4i __attribute__((vector_size(16)));
typedef __attribute__((address_space(1))) v4i gv4i;   // global-AS 16B vector
typedef __attribute__((address_space(3))) v4i lv4i;   // LDS-AS 16B vector
#define ASYNC_G2L_B128(gptr, lptr) \
  __builtin_amdgcn_global_load_async_to_lds_b128((gv4i*)(gptr), (lv4i*)(lptr), 0, 0)
#define ASYNC_L2G_B128(gptr, lptr) \
  __builtin_amdgcn_global_store_async_from_lds_b128((gv4i*)(gptr), (lv4i*)(lptr), 0, 0)
#if __has_builtin(__builtin_amdgcn_s_wait_asynccnt)
#define WAIT_ASYNC(n) __builtin_amdgcn_s_wait_asynccnt(n)
#else
#define WAIT_ASYNC(n) asm volatile("s_wait_asynccnt %0" ::"i"(n) : "memory")
#endif
#define WAIT_DS0() asm volatile("s_wait_dscnt 0x0" ::: "memory")
#endif

constexpr int HH    = 256;
constexpr int WW    = 256;
constexpr int NKY   = 12;            // ky modes kept
constexpr int NKX   = 24;            // kx rows kept (0..11 and 244..255)
constexpr int NMODE = NKX * NKY;     // 288
constexpr float TWO_PI = 6.283185307179586f;

// ---------------------------------------------------------------------------
// Kernel 1: forward partial DFT.  One block per (b,ci) image, 256 threads.
// ---------------------------------------------------------------------------
__global__ void fno_fwd_dft(const float* __restrict__ x, float* __restrict__ xft) {
  extern __shared__ float smem[];
  float* fwT = smem;                  // [32 cols][260 w]  transposed DFT matrix
  float* twp = fwT + 32 * 260;        // [256][2] interleaved (cos,sin)
  float* xs  = twp + 512;             // [8 waves][2 bufs][16 rows][68] staged x
  float* tmp = xs  + 8 * 2 * 16 * 68; // [256 h][34] row-DFT result (b64-aligned)

  const int tid  = threadIdx.x;
  const int wave = tid >> 5;
  const int lane = tid & 31;
  const int img  = blockIdx.x;                       // b*32 + ci
  const float* xim = x + (size_t)img * (HH * WW);

  // fwT[c][w]: c=2*ky -> cos(2pi ky w/256), c=2*ky+1 -> -sin(...); cols 24..31 = 0.
  for (int idx = tid; idx < 32 * 256; idx += 256) {
    int c = idx >> 8, w = idx & 255;
    float v = 0.0f;
    if (c < 2 * NKY) {
      int ky = c >> 1;
      float ang = TWO_PI * (float)((ky * w) & 255) * (1.0f / 256.0f);
      v = (c & 1) ? -__sinf(ang) : __cosf(ang);
    }
    fwT[c * 260 + w] = v;
  }
  {
    int m = tid;
    float ang = TWO_PI * (float)m * (1.0f / 256.0f);
    twp[2 * m]     = __cosf(ang);
    twp[2 * m + 1] = __sinf(ang);
  }
  __syncthreads();

  // Stage-1 GEMM: tmp[256,24] = x[256,256] @ Fw[256,24(+8 pad)].
  float* buf0 = xs + wave * (2 * 16 * 68);
  float* buf1 = buf0 + 16 * 68;
  const int srow = (lane >> 4);           // 0/1
  const int scol = (lane & 15) * 4;
  for (int mb = wave; mb < 16; mb += 8) {
    v8f acc0 = {};
    v8f acc1 = {};
#if USE_ASYNC_LDS
    // Async DMA chunk 0 into buf0.
#pragma unroll
    for (int r2 = 0; r2 < 8; ++r2) {
      int row = r2 * 2 + srow;
      ASYNC_G2L_B128(xim + (size_t)(mb * 16 + row) * WW + scol,
                     buf0 + row * 68 + scol);
    }
#else
    float4 vbuf[8];
#pragma unroll
    for (int r2 = 0; r2 < 8; ++r2) {
      int row = r2 * 2 + srow;
      vbuf[r2] = *(const float4*)(xim + (size_t)(mb * 16 + row) * WW + scol);
    }
#endif
    for (int kc = 0; kc < 4; ++kc) {
#if USE_ASYNC_LDS
      float* cur = (kc & 1) ? buf1 : buf0;
      if (kc < 3) {                       // async-prefetch next chunk
        float* nxt = (kc & 1) ? buf0 : buf1;
#pragma unroll
        for (int r2 = 0; r2 < 8; ++r2) {
          int row = r2 * 2 + srow;
          ASYNC_G2L_B128(xim + (size_t)(mb * 16 + row) * WW + (kc + 1) * 64 + scol,
                         nxt + row * 68 + scol);
        }
        WAIT_ASYNC(8);                    // chunk kc landed (in-order completion)
      } else {
        WAIT_ASYNC(0);
      }
#else
      float* cur = buf0;
#pragma unroll
      for (int r2 = 0; r2 < 8; ++r2) {    // commit prefetched chunk to LDS
        int row = r2 * 2 + srow;
        *(float4*)(cur + row * 68 + scol) = vbuf[r2];
      }
      if (kc < 3) {
#pragma unroll
        for (int r2 = 0; r2 < 8; ++r2) {  // prefetch next chunk
          int row = r2 * 2 + srow;
          vbuf[r2] = *(const float4*)(xim + (size_t)(mb * 16 + row) * WW +
                                      (kc + 1) * 64 + scol);
        }
      }
#endif
#pragma unroll
      for (int ks = 0; ks < 16; ++ks) {   // 16 WMMA k-steps (K=4 each)
        int k4 = ks * 4;
        int kglob = kc * 64 + k4;
        // A fragment: lanes 0-15 hold K=k,k+1 ; lanes 16-31 hold K=k+2,k+3.
        int arow = lane & 15;
        int acol = k4 + (srow << 1);
        v2f a = *(const v2f*)(cur + arow * 68 + acol);
        // B fragments from transposed table: one ds_load_b64 each.
        int bn   = lane & 15;
        int brow = kglob + (srow << 1);
        v2f b0 = *(const v2f*)(fwT + bn * 260 + brow);
        v2f b1 = *(const v2f*)(fwT + (16 + bn) * 260 + brow);
        acc0 = WMMA_F32(a, b0, acc0);
        acc1 = WMMA_F32(a, b1, acc1);
      }
    }
    // Spill C tiles to tmp[h][34].
    int rowoff = (lane < 16) ? 0 : 8;
    int coloff = lane & 15;
#pragma unroll
    for (int r = 0; r < 8; ++r) {
      int h = mb * 16 + rowoff + r;
      tmp[h * 34 + coloff]      = acc0[r];
      tmp[h * 34 + 16 + coloff] = acc1[r];
    }
  }
  __syncthreads();

  // Stage-2: 256-point column DFT -> 24x12 complex modes (b64 LDS reads).
  for (int o = tid; o < NMODE; o += 256) {
    int j  = o / NKY, ky = o % NKY;
    int kx = (j < 12) ? j : (232 + j);             // 244..255 for high corner
    float xr = 0.0f, xi = 0.0f;
    for (int h = 0; h < 256; ++h) {
      v2f t  = *(const v2f*)(tmp + h * 34 + 2 * ky);
      v2f cs = *(const v2f*)(twp + (((kx * h) & 255) << 1));  // e^{-i th}=c-is
      xr += t.x * cs.x + t.y * cs.y;
      xi += t.y * cs.x - t.x * cs.y;
    }
    float* dst = xft + ((size_t)img * NMODE + o) * 2;
    dst[0] = xr;
    dst[1] = xi;
  }
}

// ---------------------------------------------------------------------------
// Kernel 2: per-mode complex channel mixing.  One block per mode, 128 threads.
// ---------------------------------------------------------------------------
__global__ void fno_mix(const float* __restrict__ xft, const float* __restrict__ w1,
                        const float* __restrict__ w2, float* __restrict__ oft) {
  extern __shared__ float smem[];
  float* Ar   = smem;               // [16 b][36 i]
  float* Ai   = Ar   + 16 * 36;
  float* BrT  = Ai   + 16 * 36;     // [32 o][36 i]  (transposed for b64 B frags)
  float* BiT  = BrT  + 32 * 36;
  float* BniT = BiT  + 32 * 36;     // -Bi (f32 WMMA has no A/B negate)

  const int tid = threadIdx.x, wave = tid >> 5, lane = tid & 31;
  const int mode = blockIdx.x;                    // 0..287
  const int j = mode / NKY, ky = mode % NKY;
  const float* wsel = (j < 12) ? w1 : w2;
  const int r = (j < 12) ? j : (j - 12);

  for (int idx = tid; idx < 16 * 32; idx += 128) {
    int b = idx >> 5, i = idx & 31;
    const float* src = xft + ((size_t)(b * 32 + i) * NMODE + mode) * 2;
    Ar[b * 36 + i] = src[0];
    Ai[b * 36 + i] = src[1];
  }
  for (int idx = tid; idx < 32 * 32; idx += 128) {
    int i = idx >> 5, o = idx & 31;
    const float* src = wsel + (((size_t)(i * 32 + o) * 12 + r) * 12 + ky) * 2;
    float wr = src[0], wi = src[1];
    BrT [o * 36 + i] = wr;
    BiT [o * 36 + i] = wi;
    BniT[o * 36 + i] = -wi;
  }
  __syncthreads();

  // wave 0: Re tile0, wave 1: Re tile1, wave 2: Im tile0, wave 3: Im tile1.
  const int nt   = wave & 1;
  const int isIm = wave >> 1;
  const float* B1 = isIm ? BiT : BrT;   // paired with Xr
  const float* B2 = isIm ? BrT : BniT;  // paired with Xi
  v8f acc = {};
#pragma unroll
  for (int k = 0; k < 32; k += 4) {
    int arow = lane & 15;
    int acol = k + ((lane >> 4) << 1);
    v2f a1 = *(const v2f*)(Ar + arow * 36 + acol);
    v2f a2 = *(const v2f*)(Ai + arow * 36 + acol);
    int bn   = nt * 16 + (lane & 15);
    int brow = k + ((lane >> 4) << 1);
    v2f b1 = *(const v2f*)(B1 + bn * 36 + brow);
    v2f b2 = *(const v2f*)(B2 + bn * 36 + brow);
    acc = WMMA_F32(a1, b1, acc);
    acc = WMMA_F32(a2, b2, acc);
  }
  const int b0 = (lane < 16) ? 0 : 8;
  const int oc = nt * 16 + (lane & 15);
#pragma unroll
  for (int rr = 0; rr < 8; ++rr) {
    int b = b0 + rr;
    oft[(((size_t)(b * 32 + oc)) * NMODE + mode) * 2 + isIm] = acc[rr];
  }
}

// ---------------------------------------------------------------------------
// Kernel 3: inverse.  One block per (b,co) image, 256 threads.
// ---------------------------------------------------------------------------
__global__ void fno_inv(const float* __restrict__ oft, float* __restrict__ out) {
  extern __shared__ float smem[];
  float* GT  = smem;                 // [256 w][26 c]  transposed inverse-DFT matrix
  float* T   = GT  + 256 * 26;       // [256 h][26 c]
  float* twp = T   + 256 * 26;       // [256][2] interleaved (cos,sin)
  float* Xs  = twp + 512;            // [288][2]
  float* st  = Xs  + NMODE * 2;      // [8 waves][2 bufs][16 rows][68] staging

  const int tid = threadIdx.x, wave = tid >> 5, lane = tid & 31;
  const int img = blockIdx.x;                      // b*32 + co

  // GT[w][2*ky]=cos(2pi ky w/256), GT[w][2*ky+1]=-sin(...).
  for (int idx = tid; idx < 24 * 256; idx += 256) {
    int c = idx >> 8, w = idx & 255;
    int ky = c >> 1;
    float ang = TWO_PI * (float)((ky * w) & 255) * (1.0f / 256.0f);
    GT[w * 26 + c] = (c & 1) ? -__sinf(ang) : __cosf(ang);
  }
  {
    int m = tid;
    float ang = TWO_PI * (float)m * (1.0f / 256.0f);
    twp[2 * m]     = __cosf(ang);
    twp[2 * m + 1] = __sinf(ang);
  }
  for (int idx = tid; idx < NMODE * 2; idx += 256)
    Xs[idx] = oft[(size_t)img * NMODE * 2 + idx];
  __syncthreads();

  // Phase 1: inverse DFT along kx (24 terms); fold 1/(H*W) and the x2 for
  // ky>0; drop the DC imaginary column exactly as irfft does.
  {
    const int h = tid;
    for (int ky = 0; ky < NKY; ++ky) {
      float tr = 0.0f, ti = 0.0f;
      for (int jj = 0; jj < NKX; ++jj) {
        int kx = (jj < 12) ? jj : (232 + jj);
        v2f X  = *(const v2f*)(Xs + (jj * NKY + ky) * 2);
        v2f cs = *(const v2f*)(twp + (((kx * h) & 255) << 1));  // e^{+i th}
        tr += X.x * cs.x - X.y * cs.y;
        ti += X.x * cs.y + X.y * cs.x;
      }
      float scale = ((ky == 0) ? 1.0f : 2.0f) * (1.0f / 65536.0f);
      T[h * 26 + 2 * ky]     = scale * tr;
      T[h * 26 + 2 * ky + 1] = (ky == 0) ? 0.0f : scale * ti;
    }
  }
  __syncthreads();

  // Phase 2: out(256x256) = T(256x24) @ G(24x256).  A fragments hoisted across
  // the 16 N tiles; C tiles staged in (double-buffered) LDS and streamed out
  // with async LDS->global stores when available.
  float* outim = out + (size_t)img * (HH * WW);
  float* stb0 = st + wave * (2 * 16 * 68);
  float* stb1 = stb0 + 16 * 68;
  const int srow = lane >> 4;
  const int scol = (lane & 15) * 4;
  int g = 0;                                 // group counter (8 per wave)
  for (int mt2 = 0; mt2 < 2; ++mt2) {
    const int mt = wave * 2 + mt2;
    v2f afr[6];
#pragma unroll
    for (int kk = 0; kk < 6; ++kk) {
      int arow = mt * 16 + (lane & 15);
      int acol = kk * 4 + (srow << 1);
      afr[kk] = *(const v2f*)(T + arow * 26 + acol);
    }
    for (int ng = 0; ng < 4; ++ng, ++g) {    // groups of 4 N tiles -> 64 cols
      float* stw = (g & 1) ? stb1 : stb0;
#if USE_ASYNC_LDS
      if (g >= 2) WAIT_ASYNC(8);             // this buffer's old stores drained
#endif
#pragma unroll
      for (int q = 0; q < 4; ++q) {
        int ntl = ng * 4 + q;
        v8f acc = {};
#pragma unroll
        for (int kk = 0; kk < 6; ++kk) {
          int bn   = ntl * 16 + (lane & 15);
          int brow = kk * 4 + (srow << 1);
          v2f b = *(const v2f*)(GT + bn * 26 + brow);
          acc = WMMA_F32(afr[kk], b, acc);
        }
        int rowoff = (lane < 16) ? 0 : 8;
#pragma unroll
        for (int rr = 0; rr < 8; ++rr)
          stw[(rowoff + rr) * 68 + q * 16 + (lane & 15)] = acc[rr];
      }
#if USE_ASYNC_LDS
      WAIT_DS0();                            // spills visible to async engine
#pragma unroll
      for (int r2 = 0; r2 < 8; ++r2) {       // async LDS -> global (256B rows)
        int row = r2 * 2 + srow;
        ASYNC_L2G_B128(outim + (size_t)(mt * 16 + row) * WW + ng * 64 + scol,
                       stw + row * 68 + scol);
      }
#else
#pragma unroll
      for (int r2 = 0; r2 < 8; ++r2) {
        int row = r2 * 2 + srow;
        float4 vv = *(const float4*)(stw + row * 68 + scol);
        *(float4*)(outim + (size_t)(mt * 16 + row) * WW + ng * 64 + scol) = vv;
      }
#endif
    }
  }
  // S_ENDPGM performs an implicit wait-idle, draining outstanding async stores.
}

// ---------------------------------------------------------------------------
extern "C" void kernel_launch(void* const* d_in, const int* in_sizes, int n_in,
                              void* d_out, int out_size, void* d_ws, size_t ws_size,
                              hipStream_t stream) {
  const float* x  = (const float*)d_in[0];   // [16,32,256,256]
  const float* w1 = (const float*)d_in[1];   // [32,32,12,12,2]
  const float* w2 = (const float*)d_in[2];   // [32,32,12,12,2]
  float* out = (float*)d_out;                // [16,32,256,256]

  float* xft = (float*)d_ws;                 // [512][288][2]
  float* oft = xft + (size_t)512 * NMODE * 2;

  const size_t smem_fwd = (size_t)(32 * 260 + 512 + 8 * 2 * 16 * 68 + 256 * 34) * sizeof(float);
  const size_t smem_mix = (size_t)(2 * 16 * 36 + 3 * 32 * 36) * sizeof(float);
  const size_t smem_inv = (size_t)(256 * 26 + 256 * 26 + 512 + NMODE * 2 + 8 * 2 * 16 * 68) * sizeof(float);

  fno_fwd_dft<<<512, 256, smem_fwd, stream>>>(x, xft);
  fno_mix   <<<NMODE, 128, smem_mix, stream>>>(xft, w1, w2, oft);
  fno_inv   <<<512, 256, smem_inv, stream>>>(oft, out);
}